// GFEModule_76124000354393
// MI455X (gfx1250) — compile-verified
//
#include <hip/hip_runtime.h>

typedef float v2f __attribute__((ext_vector_type(2)));
typedef float v8f __attribute__((ext_vector_type(8)));

#define BN    128
#define BB    4
#define NF    32
#define CCH   256
#define SS    16
#define HW    256
#define TOKEN 65536
#define DD    512
#define KSPLIT 64
#define KCHUNK (TOKEN / KSPLIT)   // 1024

// ---------------------------------------------------------------------------
// tp init: tp[m, n] = bp[n]
// ---------------------------------------------------------------------------
__global__ void k_init_tp(float* __restrict__ tp, const float* __restrict__ bp) {
  int i = blockIdx.x * blockDim.x + threadIdx.x;  // 65536
  tp[i] = bp[i & (DD - 1)];
}

// ---------------------------------------------------------------------------
// Split-K WMMA GEMM: tp += tokens(128xTOKEN) @ Wp(512xTOKEN)^T
// grid (8, 32, KSPLIT), 32 threads (one wave per 16x16 tile / K-chunk)
// ---------------------------------------------------------------------------
__global__ void __launch_bounds__(32) k_gemm_tp_splitk(
    const float* __restrict__ tokens, const float* __restrict__ Wp,
    float* __restrict__ tp) {
  const int m0 = blockIdx.x * 16;
  const int n0 = blockIdx.y * 16;
  const int k0 = blockIdx.z * KCHUNK;
  const int lane = threadIdx.x;
  const int half = lane >> 4;
  const int lm = lane & 15;

  const float* ap = tokens + (size_t)(m0 + lm) * TOKEN + 2 * half;
  const float* wp = Wp + (size_t)(n0 + lm) * TOKEN + 2 * half;

  v8f acc = {0.f, 0.f, 0.f, 0.f, 0.f, 0.f, 0.f, 0.f};
  for (int k = k0; k < k0 + KCHUNK; k += 4) {
    v2f a = *(const v2f*)(ap + k);   // A[M=lm][K=k+2h .. +1]
    v2f b = *(const v2f*)(wp + k);   // B[K=k+2h .. +1][N=lm] = Wp[n][k]
    acc = __builtin_amdgcn_wmma_f32_16x16x4_f32(false, a, false, b,
                                                (short)0, acc, false, false);
  }
  const int col = n0 + lm;
#pragma unroll
  for (int j = 0; j < 8; ++j) {
    int row = m0 + j + 8 * half;
    atomicAdd(&tp[(size_t)row * DD + col], acc[j]);
  }
}

// ---------------------------------------------------------------------------
// Generic WMMA GEMM:  C[m,n] = sum_k A[g(m),k] * W[n,k] + bias[n] (+ add[m,n])
// grid (M/16, N/16), 32 threads. gather_next: row -> (b, min(n+1,31))
// ---------------------------------------------------------------------------
__global__ void __launch_bounds__(32) k_gemm_bt(
    const float* __restrict__ A, int lda,
    const float* __restrict__ W, int ldw,
    const float* __restrict__ bias,
    const float* __restrict__ addsrc,
    float* __restrict__ Cd, int ldc, int K, int gather_next) {
  const int m0 = blockIdx.x * 16;
  const int n0 = blockIdx.y * 16;
  const int lane = threadIdx.x;
  const int half = lane >> 4;
  const int lm = lane & 15;

  int rowA = m0 + lm;
  if (gather_next) {
    int b = rowA >> 5, n = rowA & 31;
    int nn = n + 1; if (nn > NF - 1) nn = NF - 1;
    rowA = (b << 5) + nn;
  }
  const float* ap = A + (size_t)rowA * lda + 2 * half;
  const float* wp = W + (size_t)(n0 + lm) * ldw + 2 * half;

  v8f acc = {0.f, 0.f, 0.f, 0.f, 0.f, 0.f, 0.f, 0.f};
  for (int k = 0; k < K; k += 4) {
    v2f a = *(const v2f*)(ap + k);
    v2f b = *(const v2f*)(wp + k);
    acc = __builtin_amdgcn_wmma_f32_16x16x4_f32(false, a, false, b,
                                                (short)0, acc, false, false);
  }
  const int col = n0 + lm;
  const float bv = bias ? bias[col] : 0.f;
#pragma unroll
  for (int j = 0; j < 8; ++j) {
    int row = m0 + j + 8 * half;
    float v = acc[j] + bv;
    if (addsrc) v += addsrc[(size_t)row * ldc + col];
    Cd[(size_t)row * ldc + col] = v;
  }
}

// ---------------------------------------------------------------------------
// Per-image channel mix:  Y[img,d,hw] = sum_c W[d,c]*X[img,c,hw] + bias[d]
// optional relu, optional residual add. grid (16, 16, 128), 32 threads.
// ---------------------------------------------------------------------------
__global__ void __launch_bounds__(32) k_gemm_chan(
    const float* __restrict__ X, const float* __restrict__ Wm,
    const float* __restrict__ bias, const float* __restrict__ addsrc,
    float* __restrict__ Y, int relu) {
  const int m0 = blockIdx.x * 16;   // out channel d
  const int n0 = blockIdx.y * 16;   // spatial hw
  const int img = blockIdx.z;
  const int lane = threadIdx.x;
  const int half = lane >> 4;
  const int lm = lane & 15;

  const float* xb = X + (size_t)img * TOKEN;
  const float* ap = Wm + (size_t)(m0 + lm) * CCH + 2 * half;

  v8f acc = {0.f, 0.f, 0.f, 0.f, 0.f, 0.f, 0.f, 0.f};
  for (int k = 0; k < CCH; k += 4) {
    v2f a = *(const v2f*)(ap + k);
    int kb = k + 2 * half;
    v2f b;
    b.x = xb[(size_t)kb * HW + n0 + lm];
    b.y = xb[(size_t)(kb + 1) * HW + n0 + lm];
    acc = __builtin_amdgcn_wmma_f32_16x16x4_f32(false, a, false, b,
                                                (short)0, acc, false, false);
  }
  const int col = n0 + lm;
#pragma unroll
  for (int j = 0; j < 8; ++j) {
    int row = m0 + j + 8 * half;
    float v = acc[j] + bias[row];
    if (relu) v = fmaxf(v, 0.f);
    size_t off = (size_t)img * TOKEN + (size_t)row * HW + col;
    if (addsrc) v += addsrc[off];
    Y[off] = v;
  }
}

// ---------------------------------------------------------------------------
// Row LayerNorm over D=512 (in place):  x = (x-mu)*rsqrt(var+eps)*g + b
// 128 blocks x 256 threads
// ---------------------------------------------------------------------------
__global__ void k_ln_rows(float* __restrict__ X, const float* __restrict__ g,
                          const float* __restrict__ b) {
  __shared__ float s1[256], s2[256];
  const int row = blockIdx.x, t = threadIdx.x;
  float* xr = X + (size_t)row * DD;
  float x0 = xr[t], x1 = xr[t + 256];
  s1[t] = x0 + x1;
  s2[t] = x0 * x0 + x1 * x1;
  __syncthreads();
  for (int o = 128; o > 0; o >>= 1) {
    if (t < o) { s1[t] += s1[t + o]; s2[t] += s2[t + o]; }
    __syncthreads();
  }
  float mu = s1[0] * (1.f / DD);
  float var = s2[0] * (1.f / DD) - mu * mu;
  float rs = rsqrtf(var + 1e-5f);
  xr[t]       = (x0 - mu) * rs * g[t]       + b[t];
  xr[t + 256] = (x1 - mu) * rs * g[t + 256] + b[t + 256];
}

// ---------------------------------------------------------------------------
// Global average pool over HW: gap[bn,c]. 128 blocks x 256 threads
// ---------------------------------------------------------------------------
__global__ void k_gap(const float* __restrict__ FA, float* __restrict__ gap) {
  const int bn = blockIdx.x, c = threadIdx.x;
  const float* p = FA + (size_t)bn * TOKEN + (size_t)c * HW;
  float s = 0.f;
  for (int i = 0; i < HW; ++i) s += p[i];
  gap[bn * CCH + c] = s * (1.f / HW);
}

// hid = relu(gap @ Wk1^T + bk1). 128 blocks x 256 threads
__global__ void k_hid(const float* __restrict__ gap, const float* __restrict__ Wk1,
                      const float* __restrict__ bk1, float* __restrict__ hid) {
  const int bn = blockIdx.x, j = threadIdx.x;
  const float* g = gap + bn * CCH;
  const float* w = Wk1 + (size_t)j * CCH;
  float s = bk1[j];
  for (int c = 0; c < CCH; ++c) s += g[c] * w[c];
  hid[bn * CCH + j] = fmaxf(s, 0.f);
}

// kern = hid @ Wk2^T + bk2. grid (128, 9) x 256 threads (2304 outputs per bn)
__global__ void k_kern(const float* __restrict__ hid, const float* __restrict__ Wk2,
                       const float* __restrict__ bk2, float* __restrict__ kern) {
  const int bn = blockIdx.x;
  const int j = blockIdx.y * 256 + threadIdx.x;  // < 2304
  const float* h = hid + bn * CCH;
  const float* w = Wk2 + (size_t)j * CCH;
  float s = bk2[j];
  for (int c = 0; c < CCH; ++c) s += h[c] * w[c];
  kern[(size_t)bn * (CCH * 9) + j] = s;
}

// ---------------------------------------------------------------------------
// Per-sample per-channel 3x3 depthwise conv (zero pad).
// grid (128*256) blocks, 256 threads (one spatial pixel each).
// ---------------------------------------------------------------------------
__global__ void k_dw(const float* __restrict__ FA, const float* __restrict__ kern,
                     float* __restrict__ OUT) {
  const int bc = blockIdx.x;
  const int bn = bc >> 8, c = bc & 255;
  const int hw = threadIdx.x;
  const int h = hw >> 4, w = hw & 15;
  const float* kv = kern + (size_t)bn * (CCH * 9) + c * 9;
  const float* x = FA + (size_t)bn * TOKEN + (size_t)c * HW;
  float s = 0.f;
#pragma unroll
  for (int di = 0; di < 3; ++di)
#pragma unroll
    for (int dj = 0; dj < 3; ++dj) {
      int hh = h + di - 1, ww = w + dj - 1;
      if (hh >= 0 && hh < SS && ww >= 0 && ww < SS)
        s += kv[di * 3 + dj] * x[hh * SS + ww];
    }
  OUT[(size_t)bn * TOKEN + (size_t)c * HW + hw] = s;
}

// ---------------------------------------------------------------------------
// GroupNorm (G=32, 2048 elems per group), in place + affine per channel.
// grid 128*32 blocks x 256 threads.
// ---------------------------------------------------------------------------
__global__ void k_gn(float* __restrict__ X, const float* __restrict__ gg,
                     const float* __restrict__ gb) {
  __shared__ float s1[256], s2[256];
  const int blk = blockIdx.x;
  const int bn = blk >> 5, g = blk & 31;
  float* base = X + (size_t)bn * TOKEN + (size_t)g * 2048;
  const int t = threadIdx.x;
  float v[8], s = 0.f, sq = 0.f;
#pragma unroll
  for (int i = 0; i < 8; ++i) {
    v[i] = base[t + i * 256];
    s += v[i]; sq += v[i] * v[i];
  }
  s1[t] = s; s2[t] = sq;
  __syncthreads();
  for (int o = 128; o > 0; o >>= 1) {
    if (t < o) { s1[t] += s1[t + o]; s2[t] += s2[t + o]; }
    __syncthreads();
  }
  float mu = s1[0] * (1.f / 2048.f);
  float var = s2[0] * (1.f / 2048.f) - mu * mu;
  float rs = rsqrtf(var + 1e-5f);
#pragma unroll
  for (int i = 0; i < 8; ++i) {
    int ch = g * 8 + i;   // element e = i*256 + t  ->  channel = g*8 + e/256
    base[t + i * 256] = (v[i] - mu) * rs * gg[ch] + gb[ch];
  }
}

// ---------------------------------------------------------------------------
// Final per-sample LayerNorm over 65536 elems + dn_g/dn_b elementwise affine.
// 128 blocks x 256 threads.
// ---------------------------------------------------------------------------
__global__ void k_final_ln(const float* __restrict__ Z, const float* __restrict__ dg,
                           const float* __restrict__ db, float* __restrict__ out) {
  __shared__ float s1[256], s2[256];
  const int bn = blockIdx.x, t = threadIdx.x;
  const float* z = Z + (size_t)bn * TOKEN;
  float s = 0.f, sq = 0.f;
  for (int i = t; i < TOKEN; i += 256) {
    float v = z[i]; s += v; sq += v * v;
  }
  s1[t] = s; s2[t] = sq;
  __syncthreads();
  for (int o = 128; o > 0; o >>= 1) {
    if (t < o) { s1[t] += s1[t + o]; s2[t] += s2[t + o]; }
    __syncthreads();
  }
  float mu = s1[0] * (1.f / TOKEN);
  float var = s2[0] * (1.f / TOKEN) - mu * mu;
  float rs = rsqrtf(var + 1e-5f);
  float* o = out + (size_t)bn * TOKEN;
  for (int i = t; i < TOKEN; i += 256)
    o[i] = (z[i] - mu) * rs * dg[i] + db[i];
}

// ---------------------------------------------------------------------------
extern "C" void kernel_launch(void* const* d_in, const int* in_sizes, int n_in,
                              void* d_out, int out_size, void* d_ws, size_t ws_size,
                              hipStream_t stream) {
  (void)in_sizes; (void)n_in; (void)out_size; (void)ws_size;
  const float* feat = (const float*)d_in[0];   // (128,256,16,16) == tokens (128,65536)
  const float* Wp   = (const float*)d_in[1];
  const float* bp   = (const float*)d_in[2];
  const float* Wu   = (const float*)d_in[3];
  const float* bu   = (const float*)d_in[4];
  // d_in[5..8] = Wq,bq,Wk,bk : dead (softmax over singleton axis == 1)
  const float* Wv   = (const float*)d_in[9];
  const float* bv   = (const float*)d_in[10];
  const float* Wo   = (const float*)d_in[11];
  const float* bo   = (const float*)d_in[12];
  const float* ln1g = (const float*)d_in[13];
  const float* ln1b = (const float*)d_in[14];
  const float* Wk1  = (const float*)d_in[15];
  const float* bk1  = (const float*)d_in[16];
  const float* Wk2  = (const float*)d_in[17];
  const float* bk2  = (const float*)d_in[18];
  const float* Wpc  = (const float*)d_in[19];
  const float* bpc  = (const float*)d_in[20];
  const float* gng  = (const float*)d_in[21];
  const float* gnb  = (const float*)d_in[22];
  const float* Wf1  = (const float*)d_in[23];
  const float* bf1  = (const float*)d_in[24];
  const float* Wf2  = (const float*)d_in[25];
  const float* bf2  = (const float*)d_in[26];
  const float* dng  = (const float*)d_in[27];
  const float* dnb  = (const float*)d_in[28];

  float* ws   = (float*)d_ws;
  float* tp   = ws;                  // 128*512
  float* tv   = tp + BN * DD;        // 128*512
  float* enh  = tv + BN * DD;        // 128*512  (attn+tp, then LN'd in place)
  float* gap  = enh + BN * DD;       // 128*256
  float* hid  = gap + BN * CCH;      // 128*256
  float* kern = hid + BN * CCH;      // 128*2304
  float* bufA = kern + BN * CCH * 9; // feat_attn   (128*65536)
  float* bufB = bufA + (size_t)BN * TOKEN;   // dw / y1 scratch
  float* bufC = bufB + (size_t)BN * TOKEN;   // mix / z scratch

  // tp = tokens @ Wp^T + bp   (split-K, float atomics)
  k_init_tp<<<(BN * DD) / 256, 256, 0, stream>>>(tp, bp);
  k_gemm_tp_splitk<<<dim3(BN / 16, DD / 16, KSPLIT), 32, 0, stream>>>(feat, Wp, tp);

  // tv = tp[next] @ Wv^T + bv
  k_gemm_bt<<<dim3(BN / 16, DD / 16), 32, 0, stream>>>(tp, DD, Wv, DD, bv, nullptr,
                                                       tv, DD, DD, 1);
  // enh_pre = tv @ Wo^T + bo + tp ;  enh = LN(enh_pre)*g + b
  k_gemm_bt<<<dim3(BN / 16, DD / 16), 32, 0, stream>>>(tv, DD, Wo, DD, bo, tp,
                                                       enh, DD, DD, 0);
  k_ln_rows<<<BN, 256, 0, stream>>>(enh, ln1g, ln1b);

  // feat_attn = enh @ Wu^T + bu  -> bufA (128 x 65536)
  k_gemm_bt<<<dim3(BN / 16, TOKEN / 16), 32, 0, stream>>>(enh, DD, Wu, DD, bu,
                                                          nullptr, bufA, TOKEN, DD, 0);

  // kernel prediction branch
  k_gap<<<BN, 256, 0, stream>>>(bufA, gap);
  k_hid<<<BN, 256, 0, stream>>>(gap, Wk1, bk1, hid);
  k_kern<<<dim3(BN, 9), 256, 0, stream>>>(hid, Wk2, bk2, kern);

  // depthwise 3x3 with predicted kernels -> bufB
  k_dw<<<BN * CCH, 256, 0, stream>>>(bufA, kern, bufB);

  // channel mix Wpc + bpc -> bufC ; GroupNorm in place
  k_gemm_chan<<<dim3(CCH / 16, HW / 16, BN), 32, 0, stream>>>(bufB, Wpc, bpc,
                                                              nullptr, bufC, 0);
  k_gn<<<BN * 32, 256, 0, stream>>>(bufC, gng, gnb);

  // y1 = relu(Wf1 . feat_dc + bf1) -> bufB
  k_gemm_chan<<<dim3(CCH / 16, HW / 16, BN), 32, 0, stream>>>(bufC, Wf1, bf1,
                                                              nullptr, bufB, 1);
  // z = Wf2 . y1 + bf2 + feat_attn -> bufC
  k_gemm_chan<<<dim3(CCH / 16, HW / 16, BN), 32, 0, stream>>>(bufB, Wf2, bf2,
                                                              bufA, bufC, 0);

  // final LayerNorm over 65536 + dn affine -> d_out
  k_final_ln<<<BN, 256, 0, stream>>>(bufC, dng, dnb, (float*)d_out);
}